// InverseHaar_21612275433975
// MI455X (gfx1250) — compile-verified
//
#include <hip/hip_runtime.h>
#include <hip/hip_bf16.h>

// Inverse 2x2 Haar synthesis, MI455X (gfx1250, wave32).
//
// Bandwidth-bound: 512 MB total traffic -> ~22us at 23.3 TB/s. The 4x4
// scaled-Hadamard butterfly is executed on the matrix pipe via
// V_WMMA_F32_16X16X4_F32 (D = A(16x4) x B(4x16)):
//   A[m][k] = 0.25 * sign(m%4, k)   (per-lane constant, ISA f32-A layout:
//                                    VGPR0 = K0|K2 across lane halves, VGPR1 = K1|K3)
//   B      = 4 channel values (a,h,v,d) for 16 spatial positions
//            (same K striping as A per the ISA 8/16-bit B generalization)
// D rows replicate mod 4, so all 32 lanes end with tl,tr,bl,br for their
// column n in D[0..3]; half-waves store (tl,tr)/(bl,br) pairs to the two
// output rows as contiguous 128B b64 bursts.
// Non-temporal loads/stores: 512MB streamed once, > 192MB L2.

typedef float v2f __attribute__((ext_vector_type(2)));
typedef float v8f __attribute__((ext_vector_type(8)));

static constexpr unsigned kPlane    = 128u * 128u;        // 16384 floats per quadrant plane
static constexpr unsigned kQS       = 64u * kPlane;       // quadrant stride (a->h->v->d) in floats
static constexpr unsigned kOutPlane = 4u * kPlane;        // 65536 floats per output plane
static constexpr unsigned kChunks   = 16u * 64u * (kPlane / 16u); // 1,048,576 chunks of 16 positions

#if defined(__gfx1250__) && __has_builtin(__builtin_amdgcn_wmma_f32_16x16x4_f32)
#define IHAAR_USE_WMMA 1
#else
#define IHAAR_USE_WMMA 0
#endif

__device__ __forceinline__ float haar_sign(unsigned m, unsigned k) {
    // 2D Hadamard sign for butterfly row m (tl,tr,bl,br) and channel k (a,h,v,d),
    // with the 0.25 normalization folded in.
    return (((m & k) & 1u) ^ (((m >> 1) & (k >> 1)) & 1u)) ? -0.25f : 0.25f;
}

__global__ __launch_bounds__(256) void InverseHaar_wmma_kernel(
    const float* __restrict__ in, float* __restrict__ out) {
    const unsigned lane  = threadIdx.x & 31u;
    const unsigned n     = lane & 15u;     // column / spatial position within chunk
    const unsigned half  = lane >> 4;      // 0: K=0,1 (a,h) + top row; 1: K=2,3 (v,d) + bottom row
    const unsigned wave  = blockIdx.x * (blockDim.x >> 5) + (threadIdx.x >> 5);
    const unsigned nwave = gridDim.x * (blockDim.x >> 5);

#if IHAAR_USE_WMMA
    // A-matrix per-lane constants (ISA: lanes 0-15 hold M=0-15; VGPR0=K0|K2, VGPR1=K1|K3).
    const unsigned m4 = n & 3u;
    const unsigned k0 = half << 1;
    v2f Amat;
    Amat.x = haar_sign(m4, k0);
    Amat.y = haar_sign(m4, k0 | 1u);
#endif

    for (unsigned c = wave; c < kChunks; c += nwave) {   // wave-uniform: EXEC stays all-1s
        const unsigned pl  = c >> 10;                    // plane index b*64+g (1024 chunks/plane)
        const unsigned pin = (c & 1023u) << 4;           // position within plane (multiple of 16)
        const unsigned hrow = pin >> 7;                  // input row
        const unsigned wcol = pin & 127u;                // input col (multiple of 16)
        // input a-plane base: (b*256+g)*16384 = (pl + 192*(pl>>6)) << 14
        const unsigned inBase = ((pl + ((pl >> 6) * 192u)) << 14) + pin + n;

        // B0: a (lanes 0-15) | v (lanes 16-31); B1: h | d. Two coalesced 128B wave loads.
        const float b0 = __builtin_nontemporal_load(in + inBase + half * (2u * kQS));
        const float b1 = __builtin_nontemporal_load(in + inBase + kQS + half * (2u * kQS));

        float r0, r1;
#if IHAAR_USE_WMMA
        v2f Bmat; Bmat.x = b0; Bmat.y = b1;
        v8f Cm = {};
        // emits v_wmma_f32_16x16x4_f32
        v8f Dm = __builtin_amdgcn_wmma_f32_16x16x4_f32(
            /*neg_a=*/false, Amat, /*neg_b=*/false, Bmat,
            /*c_mod=*/(short)0, Cm, /*reuse_a=*/false, /*reuse_b=*/false);
        // D rows repeat mod 4: D[0..3] = tl,tr,bl,br for column n in every lane.
        r0 = half ? Dm[2] : Dm[0];   // tl | bl
        r1 = half ? Dm[3] : Dm[1];   // tr | br
#else
        // VALU fallback with identical memory pattern: swap halves to get all 4 channels.
        const float o0 = __shfl_xor(b0, 16, 32);
        const float o1 = __shfl_xor(b1, 16, 32);
        const float av = half ? o0 : b0;
        const float vv = half ? b0 : o0;
        const float hv = half ? o1 : b1;
        const float dv = half ? b1 : o1;
        const float s  = half ? -1.0f : 1.0f;   // top rows: +v,+d ; bottom rows: -v,-d
        r0 = 0.25f * ((av + hv) + s * (vv + dv));   // tl | bl
        r1 = 0.25f * ((av - hv) + s * (vv - dv));   // tr | br
#endif
        // Output plane is 256x256; half-waves write rows 2h (tl,tr...) and 2h+1 (bl,br...).
        const unsigned outOff = (pl << 16) + (((hrow << 1) + half) << 8) + ((wcol + n) << 1);
        v2f val; val.x = r0; val.y = r1;
        __builtin_nontemporal_store(val, reinterpret_cast<v2f*>(out + outOff));
    }
}

extern "C" void kernel_launch(void* const* d_in, const int* in_sizes, int n_in,
                              void* d_out, int out_size, void* d_ws, size_t ws_size,
                              hipStream_t stream) {
    (void)in_sizes; (void)n_in; (void)out_size; (void)d_ws; (void)ws_size;
    const float* in = reinterpret_cast<const float*>(d_in[0]);
    float* out = reinterpret_cast<float*>(d_out);
    // 4096 blocks x 256 threads = 32768 wave32s, 32 chunks each (grid-strided).
    InverseHaar_wmma_kernel<<<4096, 256, 0, stream>>>(in, out);
}